// NAFBlock_GELU_22076131901802
// MI455X (gfx1250) — compile-verified
//
#include <hip/hip_runtime.h>
#include <cmath>

typedef float v2f __attribute__((ext_vector_type(2)));
typedef float v8f __attribute__((ext_vector_type(8)));

#define BATCH 8
#define CH    256
#define HH    128
#define WW    128
#define HWSZ  (HH*WW)          // 16384
#define CHW   (CH*HWSZ)        // 4194304

__device__ __forceinline__ v8f wmma4(v2f a, v2f b, v8f c) {
  // D(16x16 f32) = A(16x4 f32) x B(4x16 f32) + C
  return __builtin_amdgcn_wmma_f32_16x16x4_f32(false, a, false, b, (short)0, c, false, false);
}

__device__ __forceinline__ float gelu1(float x) {
  return 0.5f * x * (1.0f + erff(x * 0.70710678118654752440f));
}

// ---------------------------------------------------------------------------
// Fold GroupNorm affine into the following 1x1-conv weights.
// Wf[o,c] = w[o,c]*gnw[c];  sA[o] = sum_c w[o,c]*gnb[c];  sB[o] = sum_c w[o,c]*gnw[c]
// ---------------------------------------------------------------------------
__global__ void __launch_bounds__(256) k_fold(
    const float* __restrict__ w1, const float* __restrict__ gn1w, const float* __restrict__ gn1b,
    const float* __restrict__ w4, const float* __restrict__ gn2w, const float* __restrict__ gn2b,
    float* __restrict__ Wf1, float* __restrict__ Wf4,
    float* __restrict__ s11, float* __restrict__ s12,
    float* __restrict__ s41, float* __restrict__ s42) {
  __shared__ float red[4][256];
  int o = blockIdx.x, c = threadIdx.x;
  float a1 = w1[o*CH + c], a4 = w4[o*CH + c];
  float g1w = gn1w[c], g2w = gn2w[c];
  Wf1[o*CH + c] = a1 * g1w;
  Wf4[o*CH + c] = a4 * g2w;
  red[0][c] = a1 * gn1b[c];
  red[1][c] = a1 * g1w;
  red[2][c] = a4 * gn2b[c];
  red[3][c] = a4 * g2w;
  __syncthreads();
  for (int s = 128; s > 0; s >>= 1) {
    if (c < s) {
      red[0][c] += red[0][c+s]; red[1][c] += red[1][c+s];
      red[2][c] += red[2][c+s]; red[3][c] += red[3][c+s];
    }
    __syncthreads();
  }
  if (c == 0) { s11[o] = red[0][0]; s12[o] = red[1][0]; s41[o] = red[2][0]; s42[o] = red[3][0]; }
}

// ---------------------------------------------------------------------------
// Per-block partial sum / sumsq over a per-sample chunk (deterministic).
// ---------------------------------------------------------------------------
__global__ void __launch_bounds__(256) k_gn_part(
    const float* __restrict__ src, float* __restrict__ part, int blocksPerSample) {
  __shared__ float red[2][256];
  int blk = blockIdx.x, tid = threadIdx.x;
  int b = blk / blocksPerSample, chunk = blk % blocksPerSample;
  size_t elems = (size_t)CHW / blocksPerSample;
  size_t base = (size_t)b * CHW + (size_t)chunk * elems + tid;
  int iters = (int)(elems / 256);
  float s = 0.f, ss = 0.f;
  for (int i = 0; i < iters; i++) {
    float v = src[base + (size_t)i * 256];
    s += v; ss += v * v;
  }
  red[0][tid] = s; red[1][tid] = ss;
  __syncthreads();
  for (int st = 128; st > 0; st >>= 1) {
    if (tid < st) { red[0][tid] += red[0][tid+st]; red[1][tid] += red[1][tid+st]; }
    __syncthreads();
  }
  if (tid == 0) { part[blk*2] = red[0][0]; part[blk*2+1] = red[1][0]; }
}

// ---------------------------------------------------------------------------
// Finalize GN stats per sample + effective bias:
//   beff[b,o] = bias[o] + sA[o] - mean*rstd*sB[o]
// ---------------------------------------------------------------------------
__global__ void __launch_bounds__(256) k_gn_final(
    const float* __restrict__ part, int nPart,
    const float* __restrict__ bias, const float* __restrict__ sA, const float* __restrict__ sB,
    float* __restrict__ stats, float* __restrict__ beff) {
  __shared__ float red[2][256];
  __shared__ float sh[2];
  int b = blockIdx.x, t = threadIdx.x;
  float s = 0.f, ss = 0.f;
  for (int i = t; i < nPart; i += 256) {
    s  += part[((size_t)b * nPart + i) * 2];
    ss += part[((size_t)b * nPart + i) * 2 + 1];
  }
  red[0][t] = s; red[1][t] = ss;
  __syncthreads();
  for (int st = 128; st > 0; st >>= 1) {
    if (t < st) { red[0][t] += red[0][t+st]; red[1][t] += red[1][t+st]; }
    __syncthreads();
  }
  if (t == 0) {
    float mean = red[0][0] / (float)CHW;
    float var  = red[1][0] / (float)CHW - mean * mean;
    float rstd = rsqrtf(var + 1e-5f);
    stats[b*2] = mean; stats[b*2+1] = rstd;
    sh[0] = mean; sh[1] = rstd;
  }
  __syncthreads();
  float mean = sh[0], rstd = sh[1];
  beff[b*CH + t] = bias[t] + sA[t] - mean * rstd * sB[t];
}

// ---------------------------------------------------------------------------
// Stage 1: GN1(folded) + conv1x1(w1) -> dwconv3x3 -> gelu(gelu) -> grouped1x1
//          -> write t + per-block pooled partials.
// One block per 8x8 spatial tile (10x10 with halo), all 256 channels.
// ---------------------------------------------------------------------------
__global__ void __launch_bounds__(256) k_stage1(
    const float* __restrict__ x, const float* __restrict__ Wf1,
    const float* __restrict__ stats1, const float* __restrict__ beff1,
    const float* __restrict__ w2, const float* __restrict__ b2,
    const float* __restrict__ wg1, const float* __restrict__ bg1,
    float* __restrict__ t_out, float* __restrict__ pool_part) {
  extern __shared__ float smem[];
  const int US = 101, GS = 65, TS = 65;
  float* u = smem;             // 256 x 101 (conv1 output on 10x10 halo tile; reused for tv)
  float* g = smem + 256 * US;  // 256 x 65  (gelu^2(dwconv) on 8x8 interior)

  int blk = blockIdx.x;
  int b = blk >> 8, tile = blk & 255;
  int tY = (tile >> 4) * 8, tX = (tile & 15) * 8;
  int tid = threadIdx.x, lane = tid & 31, wv = tid >> 5;
  int n = lane & 15, half = lane >> 4;
  float rstd = stats1[b*2 + 1];
  const float* xb = x + (size_t)b * CHW;
  const float* beb = beff1 + b*CH;

  // GEMM: D[m = out-channel, n = halo-pixel], K = in-channel (256)
  for (int wt = wv; wt < 112; wt += 8) {       // 16 ch-tiles x 7 px-tiles
    int mt = wt & 15, pt = wt >> 4;
    int p  = pt * 16 + n;                      // 0..111, valid < 100
    int pc = p < 100 ? p : 99;
    int hy = tY - 1 + pc / 10, wx = tX - 1 + pc % 10;
    // multiplicative mask: outside-image halo pixels must produce exactly 0 (SAME pad)
    float ivf = ((hy >= 0) & (hy < HH) & (wx >= 0) & (wx < WW)) ? 1.0f : 0.0f;
    int hc = hy < 0 ? 0 : (hy > HH-1 ? HH-1 : hy);
    int wc = wx < 0 ? 0 : (wx > WW-1 ? WW-1 : wx);
    const float* bcol = xb + (size_t)hc * WW + wc;          // + k*HWSZ
    const float* arow = Wf1 + (size_t)(mt*16 + n) * CH;     // weight row (A operand)
    v8f acc = {0.f,0.f,0.f,0.f,0.f,0.f,0.f,0.f};
#pragma unroll 8
    for (int k0 = 0; k0 < CH; k0 += 4) {
      int ka = k0 + 2*half;
      v2f a;  a.x  = arow[ka];                 a.y  = arow[ka+1];
      v2f bb; bb.x = bcol[(size_t)ka * HWSZ];  bb.y = bcol[(size_t)(ka+1) * HWSZ];
      acc = wmma4(a, bb, acc);
    }
    // batched epilogue loads (no divergent per-element load chains)
    float be[8];
#pragma unroll
    for (int r = 0; r < 8; r++) be[r] = beb[mt*16 + r + 8*half];
    if (p < 100) {
#pragma unroll
      for (int r = 0; r < 8; r++) {
        int od = mt*16 + r + 8*half;
        u[od*US + p] = ivf * (rstd * acc[r] + be[r]);
      }
    }
  }
  __syncthreads();

  // depthwise 3x3 + double GELU; thread = channel
  {
    int c = tid;
    float wt9[9];
#pragma unroll
    for (int j = 0; j < 9; j++) wt9[j] = w2[c*9 + j];
    float bias = b2[c];
    const float* uc = u + c*US;
    for (int py = 0; py < 8; py++) {
      for (int px = 0; px < 8; px++) {
        float a = bias;
#pragma unroll
        for (int dy = 0; dy < 3; dy++)
#pragma unroll
          for (int dx = 0; dx < 3; dx++)
            a += wt9[dy*3 + dx] * uc[(py+dy)*10 + (px+dx)];
        g[c*GS + py*8 + px] = gelu1(gelu1(a));
      }
    }
  }
  __syncthreads();

  // grouped 1x1 (64 groups of 4) -> store t (coalesced) + stash in LDS for pooling
  {
    int px = tid & 63, obase = (tid >> 6) * 64;
    int py = px >> 3, pxx = px & 7;
    size_t gbase = (size_t)b * CHW + (size_t)(tY + py) * WW + (tX + pxx);
    for (int j = 0; j < 64; j++) {
      int o = obase + j;
      int grp = o >> 2, oi = o & 3;
      const float* wr = wg1 + grp*16 + oi*4;
      const float* gi = g + (grp*4)*GS + px;
      float v = bg1[o] + wr[0]*gi[0] + wr[1]*gi[GS] + wr[2]*gi[2*GS] + wr[3]*gi[3*GS];
      t_out[gbase + (size_t)o * HWSZ] = v;
      u[o*TS + px] = v;                         // reuse u region as tv
    }
  }
  __syncthreads();

  // per-block pooled partials (deterministic per-channel sums)
  {
    int c = tid;
    float s = 0.f;
    for (int px = 0; px < 64; px++) s += u[c*TS + px];
    pool_part[(size_t)blk * CH + c] = s;
  }
}

// ---------------------------------------------------------------------------
// Channel attention: pooled[b,c] -> att[b,o] = b_sca[o] + sum_c pooled*w_sca[o,c]
// ---------------------------------------------------------------------------
__global__ void __launch_bounds__(256) k_att(
    const float* __restrict__ pool_part, const float* __restrict__ w_sca,
    const float* __restrict__ b_sca, float* __restrict__ att) {
  __shared__ float pooled[256];
  int b = blockIdx.x, t = threadIdx.x;
  float s = 0.f;
  for (int tile = 0; tile < 256; tile++)
    s += pool_part[((size_t)b * 256 + tile) * CH + t];
  pooled[t] = s / (float)HWSZ;
  __syncthreads();
  float a = b_sca[t];
  const float* wr = w_sca + (size_t)t * CH;
  for (int c = 0; c < CH; c++) a += pooled[c] * wr[c];
  att[b*CH + t] = a;
}

// ---------------------------------------------------------------------------
// Stage 2a: x1 = (w3 . (t*att) + b3)*beta + x   ;  fused GN2 partial stats
// ---------------------------------------------------------------------------
__global__ void __launch_bounds__(256) k_stage2a(
    const float* __restrict__ t_in, const float* __restrict__ att,
    const float* __restrict__ w3, const float* __restrict__ b3,
    const float* __restrict__ beta, const float* __restrict__ x,
    float* __restrict__ x1, float* __restrict__ part2) {
  __shared__ float att_l[256];
  __shared__ float red[2][256];
  int blk = blockIdx.x;
  int b = blk >> 7;
  int pxb = (blk & 127) * 128;
  int tid = threadIdx.x, lane = tid & 31, wv = tid >> 5;
  int n = lane & 15, half = lane >> 4;
  att_l[tid] = att[b*CH + tid];
  __syncthreads();

  float s = 0.f, ss = 0.f;
  for (int wt = wv; wt < 128; wt += 8) {       // 16 ch-tiles x 8 px-tiles
    int mt = wt & 15, pt = wt >> 4;
    int px = pxb + pt*16 + n;
    const float* bcol = t_in + (size_t)b * CHW + px;
    const float* arow = w3 + (size_t)(mt*16 + n) * CH;
    v8f acc = {0.f,0.f,0.f,0.f,0.f,0.f,0.f,0.f};
#pragma unroll 8
    for (int k0 = 0; k0 < CH; k0 += 4) {
      int ka = k0 + 2*half;
      v2f a;  a.x  = arow[ka];  a.y = arow[ka+1];
      v2f bb;
      bb.x = bcol[(size_t)ka * HWSZ]     * att_l[ka];
      bb.y = bcol[(size_t)(ka+1) * HWSZ] * att_l[ka+1];
      acc = wmma4(a, bb, acc);
    }
    // batched epilogue: bias/beta/residual loads first, then math, then stores
    int od0 = mt*16 + 8*half;
    float b3v[8], btv[8], xr[8];
#pragma unroll
    for (int r = 0; r < 8; r++) b3v[r] = b3[od0 + r];
#pragma unroll
    for (int r = 0; r < 8; r++) btv[r] = beta[od0 + r];
#pragma unroll
    for (int r = 0; r < 8; r++) xr[r] = x[(size_t)(b*CH + od0 + r) * HWSZ + px];
#pragma unroll
    for (int r = 0; r < 8; r++) {
      size_t idx = (size_t)(b*CH + od0 + r) * HWSZ + px;
      float v = (acc[r] + b3v[r]) * btv[r] + xr[r];
      x1[idx] = v;
      s += v; ss += v * v;
    }
  }
  red[0][tid] = s; red[1][tid] = ss;
  __syncthreads();
  for (int st = 128; st > 0; st >>= 1) {
    if (tid < st) { red[0][tid] += red[0][tid+st]; red[1][tid] += red[1][tid+st]; }
    __syncthreads();
  }
  if (tid == 0) { part2[blk*2] = red[0][0]; part2[blk*2+1] = red[1][0]; }
}

// ---------------------------------------------------------------------------
// Stage 2b: GN2(folded)+conv4 -> gelu(gelu) -> grouped1x1 -> conv5
//           -> out = (.)*gamma + x1.   64 pixels per block, all channels.
// ---------------------------------------------------------------------------
__global__ void __launch_bounds__(256) k_stage2b(
    const float* __restrict__ x1, const float* __restrict__ Wf4,
    const float* __restrict__ stats2, const float* __restrict__ beff2,
    const float* __restrict__ wg2, const float* __restrict__ bg2,
    const float* __restrict__ w5, const float* __restrict__ b5,
    const float* __restrict__ gamma, float* __restrict__ out) {
  extern __shared__ float smem[];   // 256 x AS
  const int AS = 72;
  int blk = blockIdx.x;
  int b = blk >> 8;
  int pxb = (blk & 255) * 64;
  int tid = threadIdx.x, lane = tid & 31, wv = tid >> 5;
  int n = lane & 15, half = lane >> 4;
  float rstd = stats2[b*2 + 1];
  const float* beb = beff2 + b*CH;

  // conv4 (GN2 folded) + gelu^2 -> LDS
  for (int wt = wv; wt < 64; wt += 8) {        // 16 ch-tiles x 4 px-tiles
    int mt = wt & 15, pt = wt >> 4;
    int px = pxb + pt*16 + n;
    const float* bcol = x1 + (size_t)b * CHW + px;
    const float* arow = Wf4 + (size_t)(mt*16 + n) * CH;
    v8f acc = {0.f,0.f,0.f,0.f,0.f,0.f,0.f,0.f};
#pragma unroll 8
    for (int k0 = 0; k0 < CH; k0 += 4) {
      int ka = k0 + 2*half;
      v2f a;  a.x  = arow[ka];                a.y  = arow[ka+1];
      v2f bb; bb.x = bcol[(size_t)ka * HWSZ]; bb.y = bcol[(size_t)(ka+1) * HWSZ];
      acc = wmma4(a, bb, acc);
    }
    int od0 = mt*16 + 8*half;
    float be[8];
#pragma unroll
    for (int r = 0; r < 8; r++) be[r] = beb[od0 + r];
#pragma unroll
    for (int r = 0; r < 8; r++) {
      float uv = rstd * acc[r] + be[r];
      smem[(od0 + r)*AS + pt*16 + n] = gelu1(gelu1(uv));
    }
  }
  __syncthreads();

  // grouped 1x1 in-place (each thread owns whole (group, pixel) cells)
  {
    int px = tid & 63;
    int g0 = (tid >> 6) * 16;
    for (int j = 0; j < 16; j++) {
      int grp = g0 + j;
      float in0 = smem[(grp*4 + 0)*AS + px];
      float in1 = smem[(grp*4 + 1)*AS + px];
      float in2 = smem[(grp*4 + 2)*AS + px];
      float in3 = smem[(grp*4 + 3)*AS + px];
      const float* wr = wg2 + grp*16;
#pragma unroll
      for (int oi = 0; oi < 4; oi++) {
        float v = bg2[grp*4 + oi] + wr[oi*4+0]*in0 + wr[oi*4+1]*in1
                                  + wr[oi*4+2]*in2 + wr[oi*4+3]*in3;
        smem[(grp*4 + oi)*AS + px] = v;
      }
    }
  }
  __syncthreads();

  // conv5 (B operand from LDS) + gamma + residual
  for (int wt = wv; wt < 64; wt += 8) {
    int mt = wt & 15, pt = wt >> 4;
    int px = pxb + pt*16 + n;
    const float* arow = w5 + (size_t)(mt*16 + n) * CH;
    const float* bcol = smem + pt*16 + n;
    v8f acc = {0.f,0.f,0.f,0.f,0.f,0.f,0.f,0.f};
#pragma unroll 8
    for (int k0 = 0; k0 < CH; k0 += 4) {
      int ka = k0 + 2*half;
      v2f a;  a.x  = arow[ka];      a.y  = arow[ka+1];
      v2f bb; bb.x = bcol[ka * AS]; bb.y = bcol[(ka+1) * AS];
      acc = wmma4(a, bb, acc);
    }
    int od0 = mt*16 + 8*half;
    float b5v[8], gmv[8], xr[8];
#pragma unroll
    for (int r = 0; r < 8; r++) b5v[r] = b5[od0 + r];
#pragma unroll
    for (int r = 0; r < 8; r++) gmv[r] = gamma[od0 + r];
#pragma unroll
    for (int r = 0; r < 8; r++) xr[r] = x1[(size_t)(b*CH + od0 + r) * HWSZ + px];
#pragma unroll
    for (int r = 0; r < 8; r++) {
      size_t idx = (size_t)(b*CH + od0 + r) * HWSZ + px;
      out[idx] = (acc[r] + b5v[r]) * gmv[r] + xr[r];
    }
  }
}

// ---------------------------------------------------------------------------
extern "C" void kernel_launch(void* const* d_in, const int* in_sizes, int n_in,
                              void* d_out, int out_size, void* d_ws, size_t ws_size,
                              hipStream_t stream) {
  (void)in_sizes; (void)n_in; (void)out_size; (void)ws_size;
  const float* x     = (const float*)d_in[0];
  const float* gn1_w = (const float*)d_in[1];
  const float* gn1_b = (const float*)d_in[2];
  const float* w1    = (const float*)d_in[3];
  const float* b1    = (const float*)d_in[4];
  const float* w2    = (const float*)d_in[5];
  const float* b2    = (const float*)d_in[6];
  const float* wg1   = (const float*)d_in[7];
  const float* bg1   = (const float*)d_in[8];
  const float* w_sca = (const float*)d_in[9];
  const float* b_sca = (const float*)d_in[10];
  const float* w3    = (const float*)d_in[11];
  const float* b3    = (const float*)d_in[12];
  const float* gn2_w = (const float*)d_in[13];
  const float* gn2_b = (const float*)d_in[14];
  const float* w4    = (const float*)d_in[15];
  const float* b4    = (const float*)d_in[16];
  const float* wg2   = (const float*)d_in[17];
  const float* bg2   = (const float*)d_in[18];
  const float* w5    = (const float*)d_in[19];
  const float* b5    = (const float*)d_in[20];
  const float* beta  = (const float*)d_in[21];
  const float* gamma = (const float*)d_in[22];

  float* outp = (float*)d_out;        // final output; doubles as scratch for `t`
  float* wsf  = (float*)d_ws;

  // workspace layout (floats)
  size_t off = 0;
  float* part1     = wsf + off; off += 2048 * 2;
  float* stats1    = wsf + off; off += 16;
  float* beff1     = wsf + off; off += BATCH * CH;
  float* Wf1       = wsf + off; off += CH * CH;
  float* s11       = wsf + off; off += CH;
  float* s12       = wsf + off; off += CH;
  float* Wf4       = wsf + off; off += CH * CH;
  float* s41       = wsf + off; off += CH;
  float* s42       = wsf + off; off += CH;
  float* pool_part = wsf + off; off += (size_t)2048 * CH;
  float* att       = wsf + off; off += BATCH * CH;
  float* part2     = wsf + off; off += 1024 * 2;
  float* stats2    = wsf + off; off += 16;
  float* beff2     = wsf + off; off += BATCH * CH;
  float* x1buf     = wsf + off; off += (size_t)BATCH * CHW;

  float* t_buf = outp;                // stage-1 output lives in d_out until stage 2b overwrites

  dim3 blk256(256);

  // 1. fold GN affines into conv1 / conv4 weights
  k_fold<<<dim3(256), blk256, 0, stream>>>(w1, gn1_w, gn1_b, w4, gn2_w, gn2_b,
                                           Wf1, Wf4, s11, s12, s41, s42);
  // 2-3. GN1 stats
  k_gn_part<<<dim3(2048), blk256, 0, stream>>>(x, part1, 256);
  k_gn_final<<<dim3(BATCH), blk256, 0, stream>>>(part1, 256, b1, s11, s12, stats1, beff1);
  // 4. stage-1 megakernel (conv1 WMMA + dwconv + gelu^2 + grouped + pooled partials)
  size_t lds1 = (size_t)(256 * 101 + 256 * 65) * sizeof(float);   // 169,984 B
  k_stage1<<<dim3(2048), blk256, lds1, stream>>>(x, Wf1, stats1, beff1, w2, b2, wg1, bg1,
                                                 t_buf, pool_part);
  // 5. attention
  k_att<<<dim3(BATCH), blk256, 0, stream>>>(pool_part, w_sca, b_sca, att);
  // 6. stage-2a (conv3 WMMA + beta + residual, fused GN2 partials)
  k_stage2a<<<dim3(1024), blk256, 0, stream>>>(t_buf, att, w3, b3, beta, x, x1buf, part2);
  // 7. GN2 finalize
  k_gn_final<<<dim3(BATCH), blk256, 0, stream>>>(part2, 128, b4, s41, s42, stats2, beff2);
  // 8. stage-2b megakernel (conv4 WMMA + gelu^2 + grouped + conv5 WMMA + gamma + residual)
  size_t lds2 = (size_t)(256 * 72) * sizeof(float);               // 73,728 B
  k_stage2b<<<dim3(2048), blk256, lds2, stream>>>(x1buf, Wf4, stats2, beff2, wg2, bg2,
                                                  w5, b5, gamma, outp);
}